// MinamoSimilarityTopo_38079180047101
// MI455X (gfx1250) — compile-verified
//
#include <hip/hip_runtime.h>

namespace {
constexpr int kNodes = 20000;
constexpr int kEdges = 320000;
constexpr int kBatch = 16;
}

typedef __attribute__((ext_vector_type(2))) float v2f;
typedef __attribute__((ext_vector_type(8))) float v8f;

// ---------------------------------------------------------------- utilities
__global__ __launch_bounds__(256) void k_fill0(float* __restrict__ p, int n) {
  int i = blockIdx.x * 256 + threadIdx.x;
  if (i < n) p[i] = 0.0f;
}

__global__ __launch_bounds__(256) void k_deg(const int* __restrict__ dst,
                                             float* __restrict__ deg) {
  int e = blockIdx.x * 256 + threadIdx.x;
  if (e < kEdges) atomicAdd(&deg[dst[e]], 1.0f);
}

__global__ __launch_bounds__(256) void k_dinv(float* __restrict__ deg) {
  int i = blockIdx.x * 256 + threadIdx.x;
  if (i < kNodes) deg[i] = rsqrtf(deg[i] + 1.0f);
}

__global__ __launch_bounds__(256) void k_coef(const int* __restrict__ src,
                                              const int* __restrict__ dst,
                                              const float* __restrict__ dinv,
                                              float* __restrict__ coef) {
  int e = blockIdx.x * 256 + threadIdx.x;
  if (e < kEdges) coef[e] = dinv[src[e]] * dinv[dst[e]];
}

// ------------------------------------------------------------- fp32 WMMA GEMM
// C[M,N] = A[M,K] @ W[K,N] (+ bias). One workgroup per 16-row tile:
//  - stage the 16xK A panel into LDS (coalesced float4, row stride K+4 so the
//    32-lane ds_load_b64 A-fragment fetch is bank-conflict free),
//  - each wave owns TN 16-col tiles -> TN independent WMMA accumulation chains
//    sharing one A fragment per k-step.
// WMMA operand layouts per CDNA5 ISA:
//  A 16x4 : lanes 0-15 hold (k,k+1) of row M=lane; lanes 16-31 hold (k+2,k+3).
//  B 4x16 : VGPR0 = rows k / k+2 split across lane halves, VGPR1 = k+1 / k+3.
//  C/D    : VGPR j = row row0 + j + (lane>=16 ? 8 : 0), col = lane%16.
template <int K, int N, int WAVES, int TN>
__global__ __launch_bounds__(WAVES * 32) void k_gemm_wmma(
    const float* __restrict__ A, const float* __restrict__ Wm,
    const float* __restrict__ bias, float* __restrict__ C) {
  static_assert(N == WAVES * TN * 16, "tile coverage");
  constexpr int LDA = K + 4;  // pad: lane l reads banks {4l+koff, 4l+koff+1}
  __shared__ float sA[16 * LDA];

  const int row0 = blockIdx.x << 4;
  const int tid  = threadIdx.x;

  // cooperative, coalesced stage of the 16xK A row-panel into LDS
  constexpr int NV4 = (16 * K) / 4;
#pragma unroll 2
  for (int i = tid; i < NV4; i += WAVES * 32) {
    int r  = (i << 2) / K;
    int cf = (i << 2) - r * K;
    float4 v = *(const float4*)(A + (size_t)(row0 + r) * K + cf);
    *(float4*)(&sA[r * LDA + cf]) = v;
  }
  __syncthreads();

  const int wave = tid >> 5;
  const int lane = tid & 31;
  const int l16  = lane & 15;
  const int koff = (lane >> 4) << 1;  // 0 for lanes 0-15, 2 for lanes 16-31

  const float* aBase = &sA[l16 * LDA + koff];
  const float* wBase = Wm + (size_t)koff * N + l16;

  v8f acc[TN] = {};

#pragma unroll 4
  for (int k = 0; k < K; k += 4) {
    v2f a = *(const v2f*)(aBase + k);  // ds_load_b64, conflict-free
#pragma unroll
    for (int j = 0; j < TN; ++j) {
      const int col0 = (wave * TN + j) << 4;
      v2f b;
      b.x = wBase[(size_t)k * N + col0];        // immediate-offset global loads
      b.y = wBase[(size_t)(k + 1) * N + col0];
      acc[j] = __builtin_amdgcn_wmma_f32_16x16x4_f32(
          /*neg_a=*/false, a, /*neg_b=*/false, b,
          /*c_mod=*/(short)0, acc[j], /*reuse_a=*/false, /*reuse_b=*/false);
    }
  }

  const int rbase = row0 + ((lane >> 4) << 3);
#pragma unroll
  for (int j = 0; j < TN; ++j) {
    const int col = ((wave * TN + j) << 4) + l16;
    float bv = bias ? bias[col] : 0.0f;
#pragma unroll
    for (int r = 0; r < 8; ++r)
      C[(size_t)(rbase + r) * N + col] = acc[j][r] + bv;
  }
}

// ------------------------------------------------------ LayerNorm + ReLU (row)
__global__ __launch_bounds__(32) void k_ln_relu(const float* __restrict__ X,
                                                const float* __restrict__ g,
                                                const float* __restrict__ b,
                                                float* __restrict__ H, int F) {
  int row  = blockIdx.x;
  int lane = threadIdx.x;
  const float* x = X + (size_t)row * F;
  float s = 0.0f, s2 = 0.0f;
  for (int i = lane; i < F; i += 32) {
    float v = x[i];
    s += v;
    s2 += v * v;
  }
#pragma unroll
  for (int o = 16; o > 0; o >>= 1) {
    s  += __shfl_xor(s, o, 32);
    s2 += __shfl_xor(s2, o, 32);
  }
  float inv = 1.0f / (float)F;
  float mu  = s * inv;
  float var = s2 * inv - mu * mu;
  float rs  = rsqrtf(var + 1e-5f);
  float* h  = H + (size_t)row * F;
  for (int i = lane; i < F; i += 32) {
    float v = (x[i] - mu) * rs * g[i] + b[i];
    h[i] = v > 0.0f ? v : 0.0f;
  }
}

// ------------------------------------------------- edge gather/scatter message
// thread per (edge, 4 features); fshift = log2(F/4)
__global__ __launch_bounds__(256) void k_scatter(const float* __restrict__ hW,
                                                 const int* __restrict__ src,
                                                 const int* __restrict__ dst,
                                                 const float* __restrict__ coef,
                                                 float* __restrict__ agg, int fshift) {
  int idx = blockIdx.x * 256 + threadIdx.x;
  if (idx >= (kEdges << fshift)) return;
  int e  = idx >> fshift;
  int f4 = (idx & ((1 << fshift) - 1)) << 2;
  int F  = 4 << fshift;
  float  c = coef[e];
  float4 v = *(const float4*)(hW + (size_t)src[e] * F + f4);
  float* o = agg + (size_t)dst[e] * F + f4;
  atomicAdd(o + 0, v.x * c);
  atomicAdd(o + 1, v.y * c);
  atomicAdd(o + 2, v.z * c);
  atomicAdd(o + 3, v.w * c);
}

// agg += hW * dinv^2 + bias   (self-loop term of symmetric-normalized A+I)
__global__ __launch_bounds__(256) void k_selfloop_bias(const float* __restrict__ hW,
                                                       const float* __restrict__ dinv,
                                                       const float* __restrict__ bias,
                                                       float* __restrict__ agg, int fshift) {
  int idx = blockIdx.x * 256 + threadIdx.x;
  if (idx >= (kNodes << fshift)) return;
  int n  = idx >> fshift;
  int f4 = (idx & ((1 << fshift) - 1)) << 2;
  int F  = 4 << fshift;
  float d  = dinv[n];
  float d2 = d * d;
  float4 v = *(const float4*)(hW + (size_t)n * F + f4);
  float4 a = *(float4*)(agg + (size_t)n * F + f4);
  a.x += v.x * d2 + bias[f4 + 0];
  a.y += v.y * d2 + bias[f4 + 1];
  a.z += v.z * d2 + bias[f4 + 2];
  a.w += v.w * d2 + bias[f4 + 3];
  *(float4*)(agg + (size_t)n * F + f4) = a;
}

// ------------------------------------------------------------------- pooling
__global__ __launch_bounds__(256) void k_pool(const float* __restrict__ h,
                                              const int* __restrict__ batch,
                                              float* __restrict__ pooled) {
  int idx = blockIdx.x * 256 + threadIdx.x;
  if (idx >= kNodes * 128) return;  // 512 feats / 4 per thread
  int n  = idx >> 7;
  int f4 = (idx & 127) << 2;
  int bg = batch[n];
  float4 v = *(const float4*)(h + (size_t)n * 512 + f4);
  float* o = pooled + (size_t)bg * 512 + f4;
  atomicAdd(o + 0, v.x);
  atomicAdd(o + 1, v.y);
  atomicAdd(o + 2, v.z);
  atomicAdd(o + 3, v.w);
}

__global__ __launch_bounds__(256) void k_count(const int* __restrict__ batch,
                                               float* __restrict__ counts) {
  int n = blockIdx.x * 256 + threadIdx.x;
  if (n < kNodes) atomicAdd(&counts[batch[n]], 1.0f);
}

__global__ __launch_bounds__(256) void k_poolavg(float* __restrict__ pooled,
                                                 const float* __restrict__ counts) {
  int idx = blockIdx.x * 256 + threadIdx.x;
  if (idx < kBatch * 512) pooled[idx] /= fmaxf(counts[idx >> 9], 1.0f);
}

// ------------------------------------------------------------------- driver
extern "C" void kernel_launch(void* const* d_in, const int* in_sizes, int n_in,
                              void* d_out, int out_size, void* d_ws, size_t ws_size,
                              hipStream_t stream) {
  (void)in_sizes; (void)n_in; (void)out_size; (void)ws_size;

  const float* x    = (const float*)d_in[0];
  const int*   ei   = (const int*)d_in[1];
  const int*   src  = ei;
  const int*   dstp = ei + kEdges;
  const int*   bat  = (const int*)d_in[2];
  const float *W0 = (const float*)d_in[3],  *b0 = (const float*)d_in[4],
              *g0 = (const float*)d_in[5],  *be0 = (const float*)d_in[6];
  const float *W1 = (const float*)d_in[7],  *b1 = (const float*)d_in[8],
              *g1 = (const float*)d_in[9],  *be1 = (const float*)d_in[10];
  const float *W2 = (const float*)d_in[11], *b2 = (const float*)d_in[12],
              *g2 = (const float*)d_in[13], *be2 = (const float*)d_in[14];
  const float *W3 = (const float*)d_in[15], *b3 = (const float*)d_in[16],
              *g3 = (const float*)d_in[17], *be3 = (const float*)d_in[18];
  const float *Wo = (const float*)d_in[19], *bo = (const float*)d_in[20];
  float* out = (float*)d_out;

  char*  ws  = (char*)d_ws;
  size_t off = 0;
  auto alloc = [&](size_t bytes) -> float* {
    float* p = (float*)(ws + off);
    off += (bytes + 255) & ~(size_t)255;
    return p;
  };
  const size_t actBytes = (size_t)kNodes * 512 * sizeof(float);
  float* h      = alloc(actBytes);
  float* hW     = alloc(actBytes);
  float* agg    = alloc(actBytes);
  float* dinv   = alloc((size_t)kNodes * sizeof(float));
  float* coef   = alloc((size_t)kEdges * sizeof(float));
  float* pooled = alloc((size_t)kBatch * 512 * sizeof(float));
  float* counts = alloc(256);

  auto cdiv = [](int a, int b) { return (a + b - 1) / b; };
  constexpr int kRowTiles = kNodes / 16;  // 1250

  // symmetric normalization coefficients (graph fixed across layers)
  k_fill0<<<cdiv(kNodes, 256), 256, 0, stream>>>(dinv, kNodes);
  k_deg<<<cdiv(kEdges, 256), 256, 0, stream>>>(dstp, dinv);
  k_dinv<<<cdiv(kNodes, 256), 256, 0, stream>>>(dinv);
  k_coef<<<cdiv(kEdges, 256), 256, 0, stream>>>(src, dstp, dinv, coef);

  // layer 0: h = relu(LN(x @ W0 + b0))
  k_gemm_wmma<32, 64, 2, 2><<<kRowTiles, 64, 0, stream>>>(x, W0, b0, h);
  k_ln_relu<<<kNodes, 32, 0, stream>>>(h, g0, be0, h, 64);

  // layer 1: Fin=64 -> Fout=128 (fshift 5)
  k_gemm_wmma<64, 128, 4, 2><<<kRowTiles, 128, 0, stream>>>(h, W1, nullptr, hW);
  k_fill0<<<cdiv(kNodes * 128, 256), 256, 0, stream>>>(agg, kNodes * 128);
  k_scatter<<<cdiv(kEdges << 5, 256), 256, 0, stream>>>(hW, src, dstp, coef, agg, 5);
  k_selfloop_bias<<<cdiv(kNodes << 5, 256), 256, 0, stream>>>(hW, dinv, b1, agg, 5);
  k_ln_relu<<<kNodes, 32, 0, stream>>>(agg, g1, be1, h, 128);

  // layer 2: Fin=128 -> Fout=256 (fshift 6)
  k_gemm_wmma<128, 256, 8, 2><<<kRowTiles, 256, 0, stream>>>(h, W2, nullptr, hW);
  k_fill0<<<cdiv(kNodes * 256, 256), 256, 0, stream>>>(agg, kNodes * 256);
  k_scatter<<<cdiv(kEdges << 6, 256), 256, 0, stream>>>(hW, src, dstp, coef, agg, 6);
  k_selfloop_bias<<<cdiv(kNodes << 6, 256), 256, 0, stream>>>(hW, dinv, b2, agg, 6);
  k_ln_relu<<<kNodes, 32, 0, stream>>>(agg, g2, be2, h, 256);

  // layer 3: Fin=256 -> Fout=512 (fshift 7)
  k_gemm_wmma<256, 512, 8, 4><<<kRowTiles, 256, 0, stream>>>(h, W3, nullptr, hW);
  k_fill0<<<cdiv(kNodes * 512, 256), 256, 0, stream>>>(agg, kNodes * 512);
  k_scatter<<<cdiv(kEdges << 7, 256), 256, 0, stream>>>(hW, src, dstp, coef, agg, 7);
  k_selfloop_bias<<<cdiv(kNodes << 7, 256), 256, 0, stream>>>(hW, dinv, b3, agg, 7);
  k_ln_relu<<<kNodes, 32, 0, stream>>>(agg, g3, be3, h, 512);

  // mean-pool per graph, then head GEMM -> d_out
  k_fill0<<<cdiv(kBatch * 512, 256), 256, 0, stream>>>(pooled, kBatch * 512);
  k_fill0<<<1, 256, 0, stream>>>(counts, kBatch);
  k_pool<<<cdiv(kNodes * 128, 256), 256, 0, stream>>>(h, bat, pooled);
  k_count<<<cdiv(kNodes, 256), 256, 0, stream>>>(bat, counts);
  k_poolavg<<<cdiv(kBatch * 512, 256), 256, 0, stream>>>(pooled, counts);

  k_gemm_wmma<512, 128, 4, 2><<<1, 128, 0, stream>>>(pooled, Wo, bo, out);
}